// SSDLayerLoss_44074954391778
// MI455X (gfx1250) — compile-verified
//
#include <hip/hip_runtime.h>
#include <math.h>

// Problem constants (from reference setup_inputs)
#define NBATCH 32
#define NPRIOR 8732
#define NCLASS 81
#define NGT    50
#define ROWS_PER_BLOCK 8                                   // 8 waves/block, 1 row per wave
#define CB ((NPRIOR + ROWS_PER_BLOCK - 1) / ROWS_PER_BLOCK) // 1092 row-tiles per batch
#define NB (NBATCH * CB)                                    // 34944 partial slots

static_assert(NB % 64 == 0, "finalize WMMA reduction assumes 64-element chunks with no tail");

typedef __attribute__((ext_vector_type(2))) float v2f;
typedef __attribute__((ext_vector_type(8))) float v8f;

// ---------------- wave32 reductions ----------------
__device__ __forceinline__ float wsum(float v) {
#pragma unroll
    for (int m = 16; m >= 1; m >>= 1) v += __shfl_xor(v, m, 32);
    return v;
}
__device__ __forceinline__ float wmax(float v) {
#pragma unroll
    for (int m = 16; m >= 1; m >>= 1) v = fmaxf(v, __shfl_xor(v, m, 32));
    return v;
}

__device__ __forceinline__ float iou_f(float gx1, float gy1, float gx2, float gy2,
                                       float px1, float py1, float px2, float py2) {
    float lx = fmaxf(gx1, px1), ly = fmaxf(gy1, py1);
    float rx = fminf(gx2, px2), ry = fminf(gy2, py2);
    float w = fmaxf(rx - lx, 0.0f), h = fmaxf(ry - ly, 0.0f);
    float inter = w * h;
    float ag = (gx2 - gx1) * (gy2 - gy1);
    float ap = (px2 - px1) * (py2 - py1);
    return inter / (ag + ap - inter);
}

// K1: per (batch, gt): argmax over priors of IoU (first-index tie-break).
__global__ void k_best_prior(const float* __restrict__ targets,
                             const float* __restrict__ priors,
                             int* __restrict__ best_prior) {
    __shared__ float sOv[256];
    __shared__ int   sIdx[256];
    int b = blockIdx.x;                 // 0 .. NBATCH*NGT-1
    int n = b / NGT, g = b % NGT;
    const float* tg = targets + ((size_t)n * NGT + g) * 6;
    float valid = tg[0];
    float gx1 = tg[2], gy1 = tg[3], gx2 = tg[4], gy2 = tg[5];
    float bOv = -3.4e38f; int bIdx = 0x7fffffff;
    for (int p = threadIdx.x; p < NPRIOR; p += 256) {
        const float4 pr = ((const float4*)priors)[p];
        float ov = (valid > 0.0f) ? iou_f(gx1, gy1, gx2, gy2, pr.x, pr.y, pr.z, pr.w) : -1.0f;
        if (ov > bOv || (ov == bOv && p < bIdx)) { bOv = ov; bIdx = p; }
    }
    sOv[threadIdx.x] = bOv; sIdx[threadIdx.x] = bIdx;
    __syncthreads();
    for (int s = 128; s >= 1; s >>= 1) {
        if (threadIdx.x < s) {
            float o2 = sOv[threadIdx.x + s]; int i2 = sIdx[threadIdx.x + s];
            if (o2 > sOv[threadIdx.x] || (o2 == sOv[threadIdx.x] && i2 < sIdx[threadIdx.x])) {
                sOv[threadIdx.x] = o2; sIdx[threadIdx.x] = i2;
            }
        }
        __syncthreads();
    }
    if (threadIdx.x == 0) best_prior[b] = sIdx[0];
}

// K2: per (batch, prior): best overlapping GT (max + first argmax over 50 GTs).
__global__ void k_best_truth(const float* __restrict__ targets,
                             const float* __restrict__ priors,
                             float* __restrict__ best_ov,
                             int* __restrict__ best_idx) {
    int t = blockIdx.x * 256 + threadIdx.x;
    if (t >= NBATCH * NPRIOR) return;
    int n = t / NPRIOR, p = t % NPRIOR;
    const float4 pr = ((const float4*)priors)[p];
    const float* tb = targets + (size_t)n * NGT * 6;
    float bOv = -3.4e38f; int bIdx = 0;
#pragma unroll 5
    for (int g = 0; g < NGT; ++g) {
        const float* tg = tb + g * 6;
        float ov = (tg[0] > 0.0f) ? iou_f(tg[2], tg[3], tg[4], tg[5], pr.x, pr.y, pr.z, pr.w) : -1.0f;
        if (ov > bOv) { bOv = ov; bIdx = g; }
    }
    best_ov[t] = bOv; best_idx[t] = bIdx;
}

// K3: scatter override: each valid GT claims its best prior (sequential per batch).
__global__ void k_override(const float* __restrict__ targets,
                           const int* __restrict__ best_prior,
                           float* __restrict__ best_ov,
                           int* __restrict__ best_idx) {
    int n = threadIdx.x;
    if (n >= NBATCH) return;
    for (int g = 0; g < NGT; ++g) {
        if (targets[((size_t)n * NGT + g) * 6] > 0.0f) {
            int bp = best_prior[n * NGT + g];
            best_ov[(size_t)n * NPRIOR + bp] = 2.0f;
            best_idx[(size_t)n * NPRIOR + bp] = g;
        }
    }
}

// K4: streaming pass. One wave32 per prior row: coalesced logsumexp over 81
// classes + lane0 does the box-encode smooth-L1. Per-block partials to ws.
__global__ void k_loss_rows(const float* __restrict__ loc_p,
                            const float* __restrict__ conf_p,
                            const float* __restrict__ priors,
                            const float* __restrict__ targets,
                            const float* __restrict__ best_ov,
                            const int* __restrict__ best_idx,
                            float* __restrict__ pnll,
                            float* __restrict__ psl1,
                            float* __restrict__ ppos) {
    __shared__ float sN[ROWS_PER_BLOCK], sL[ROWS_PER_BLOCK], sP[ROWS_PER_BLOCK];
    int lane = threadIdx.x & 31;
    int wave = threadIdx.x >> 5;
    int cb = blockIdx.x % CB;
    int n  = blockIdx.x / CB;
    int p  = cb * ROWS_PER_BLOCK + wave;
    float nll = 0.0f, sl1 = 0.0f, pos = 0.0f;
    if (p < NPRIOR) {
        const float* row = conf_p + ((size_t)n * NPRIOR + p) * NCLASS;
        float x0 = row[lane];
        float x1 = row[lane + 32];
        bool has2 = (lane + 64) < NCLASS;
        float x2 = has2 ? row[lane + 64] : -3.4e38f;
        float m = wmax(fmaxf(fmaxf(x0, x1), x2));
        float s = expf(x0 - m) + expf(x1 - m) + (has2 ? expf(x2 - m) : 0.0f);
        s = wsum(s);
        float bov = best_ov[(size_t)n * NPRIOR + p];
        int  bidx = best_idx[(size_t)n * NPRIOR + p];
        float label = targets[((size_t)n * NGT + bidx) * 6 + 1];
        float conf_t = (bov < 0.5f) ? 0.0f : label;
        int tgt = (int)conf_t;
        float xt = row[tgt];                       // uniform address -> broadcast
        nll = logf(s) + m - xt;
        if (lane == 0 && conf_t > 0.0f) {
            pos = 1.0f;
            const float4 pr = ((const float4*)priors)[p];
            float pwx = pr.z - pr.x, pwy = pr.w - pr.y;
            float pcx = (pr.x + pr.z) * 0.5f, pcy = (pr.y + pr.w) * 0.5f;
            const float* tg = targets + ((size_t)n * NGT + bidx) * 6;
            float mwx = fmaxf(tg[4] - tg[2], 1e-6f), mwy = fmaxf(tg[5] - tg[3], 1e-6f);
            float mcx = (tg[2] + tg[4]) * 0.5f,      mcy = (tg[3] + tg[5]) * 0.5f;
            float t0 = (mcx - pcx) / (0.1f * pwx);
            float t1 = (mcy - pcy) / (0.1f * pwy);
            float t2 = logf(mwx / pwx) * 5.0f;     // 1/VAR_S
            float t3 = logf(mwy / pwy) * 5.0f;
            const float* lp = loc_p + ((size_t)n * NPRIOR + p) * 4;
            float d0 = lp[0] - t0, d1 = lp[1] - t1, d2 = lp[2] - t2, d3 = lp[3] - t3;
#define SL1(d) (fabsf(d) < 1.0f ? 0.5f * (d) * (d) : fabsf(d) - 0.5f)
            sl1 = SL1(d0) + SL1(d1) + SL1(d2) + SL1(d3);
#undef SL1
        }
    }
    if (lane == 0) { sN[wave] = nll; sL[wave] = sl1; sP[wave] = pos; }
    __syncthreads();
    if (threadIdx.x == 0) {
        float a = 0.0f, b = 0.0f, c = 0.0f;
        for (int w = 0; w < ROWS_PER_BLOCK; ++w) { a += sN[w]; b += sL[w]; c += sP[w]; }
        pnll[blockIdx.x] = a; psl1[blockIdx.x] = b; ppos[blockIdx.x] = c;
    }
}

// K5: WMMA-based exact-f32 sum reduction of the three partial arrays.
// D = A(16x4) * ones(4x16) + C  =>  sum(all D entries) = 16 * sum(A).
// The lane->A-slot mapping is irrelevant because every A element contributes
// exactly once, so loads stay fully coalesced. EXEC is all-1s around the
// WMMA (wave-uniform branch), per ISA restriction.
__global__ void k_finalize(const float* __restrict__ pnll,
                           const float* __restrict__ psl1,
                           const float* __restrict__ ppos,
                           float* __restrict__ out) {
    __shared__ float res[4];
    int wave = threadIdx.x >> 5;
    int lane = threadIdx.x & 31;
    if (wave < 3) {
        const float* arr = (wave == 0) ? pnll : (wave == 1) ? psl1 : ppos;
        v8f acc = {0.0f, 0.0f, 0.0f, 0.0f, 0.0f, 0.0f, 0.0f, 0.0f};
        v2f ones; ones.x = 1.0f; ones.y = 1.0f;
        for (int base = 0; base < NB; base += 64) {
            v2f a; a.x = arr[base + lane]; a.y = arr[base + 32 + lane];
            acc = __builtin_amdgcn_wmma_f32_16x16x4_f32(
                /*neg_a=*/false, a, /*neg_b=*/false, ones,
                /*c_mod=*/(short)0, acc, /*reuse_a=*/false, /*reuse_b=*/false);
        }
        float s = acc[0] + acc[1] + acc[2] + acc[3] + acc[4] + acc[5] + acc[6] + acc[7];
        s = wsum(s) * (1.0f / 16.0f);
        if (lane == 0) res[wave] = s;
    }
    __syncthreads();
    if (threadIdx.x == 0) {
        float nllS = res[0], sl1S = res[1], posS = res[2];
        float denom = fmaxf(posS * 4.0f, 1.0f);          // max(num_pos*4, 1)
        out[0] = sl1S / denom + nllS / (float)((size_t)NBATCH * NPRIOR);
    }
}

extern "C" void kernel_launch(void* const* d_in, const int* in_sizes, int n_in,
                              void* d_out, int out_size, void* d_ws, size_t ws_size,
                              hipStream_t stream) {
    const float* loc_p   = (const float*)d_in[0];   // (32, 8732, 4)
    const float* conf_p  = (const float*)d_in[1];   // (32, 8732, 81)
    const float* priors  = (const float*)d_in[2];   // (8732, 4)
    const float* targets = (const float*)d_in[3];   // (32, 50, 6)
    float* out = (float*)d_out;                     // scalar f32 loss

    const size_t NP = (size_t)NBATCH * NPRIOR;
    float* best_ov    = (float*)d_ws;               // NP floats
    int*   best_idx   = (int*)(best_ov + NP);       // NP ints
    int*   best_prior = (int*)(best_idx + NP);      // NBATCH*NGT ints
    float* pnll       = (float*)(best_prior + NBATCH * NGT); // NB floats
    float* psl1       = pnll + NB;                  // NB floats
    float* ppos       = psl1 + NB;                  // NB floats  (~2.7 MB total)

    k_best_prior<<<NBATCH * NGT, 256, 0, stream>>>(targets, priors, best_prior);
    k_best_truth<<<(int)((NP + 255) / 256), 256, 0, stream>>>(targets, priors, best_ov, best_idx);
    k_override<<<1, 32, 0, stream>>>(targets, best_prior, best_ov, best_idx);
    k_loss_rows<<<NBATCH * CB, 256, 0, stream>>>(loc_p, conf_p, priors, targets,
                                                 best_ov, best_idx, pnll, psl1, ppos);
    k_finalize<<<1, 128, 0, stream>>>(pnll, psl1, ppos, out);
}